// Encoder_28269474742326
// MI455X (gfx1250) — compile-verified
//
#include <hip/hip_runtime.h>

typedef __attribute__((ext_vector_type(2))) float v2f;
typedef __attribute__((ext_vector_type(8))) float v8f;

#define N_NODES 50000
#define N_EDGES 1600000
#define IN_CH   256
#define HID_CH  128
#define OUT_CH  64

// ---------------------------------------------------------------------------
// Degree counting: deg[dst[e]] += 1.0 per edge
// ---------------------------------------------------------------------------
__global__ __launch_bounds__(256)
void deg_count_kernel(const int* __restrict__ dst, float* __restrict__ deg, int nE) {
    int e = blockIdx.x * 256 + threadIdx.x;
    if (e < nE) atomicAdd(&deg[dst[e]], 1.0f);
}

// dinv[i] = rsqrt(1 + count[i]), in place
__global__ __launch_bounds__(256)
void rsqrt_kernel(float* __restrict__ deg, int n) {
    int i = blockIdx.x * 256 + threadIdx.x;
    if (i < n) deg[i] = rsqrtf(deg[i] + 1.0f);
}

// ---------------------------------------------------------------------------
// FP32 WMMA GEMM with LDS staging.
//   C[M, NT*16] = A[M,K] * B[K, NT*16]  (row-major, exact fp32)
// Block = 256 threads (8 waves). Block tile: BM=128 rows x BN=NT*16 cols.
// Wave w owns rows [blockM + 16w, +16) and the full BN width (NT accumulators).
// K chunked by KB=32. A staged row-major (padded); B staged K-PAIRED:
//   BsP[(k/2)*LDB_P + n*2 + (k&1)]
// so each WMMA B operand (B[k][n], B[k+1][n]) is one aligned ds_load_b64.
//
// WMMA 16x16x4 f32 operand layouts (ISA 7.12.2):
//   A 16x4 : lanes 0-15 M=lane {K=0,1}; lanes 16-31 M=lane-16 {K=2,3}
//   B 4x16 : lanes 0-15 N=lane rows {K=0,1}; lanes 16-31 rows {K=2,3}
//   C 16x16: VGPR r -> M = r + (lane>=16 ? 8 : 0), N = lane&15
// ---------------------------------------------------------------------------
template<int NT>
__global__ __launch_bounds__(256)
void gemm_wmma_f32_kernel(const float* __restrict__ A, const float* __restrict__ B,
                          float* __restrict__ C, int M, int K,
                          int lda, int ldb, int ldc, int c_col_off) {
    constexpr int KB    = 32;             // K chunk
    constexpr int BM    = 128;            // rows per block
    constexpr int BN    = NT * 16;        // cols per block
    constexpr int LDA_S = KB + 4;         // 36 floats A row stride (16B-aligned)
    constexpr int LDB_P = 2 * BN + 8;     // paired-K row stride (16B-aligned)

    __shared__ float As [BM * LDA_S];
    __shared__ float BsP[(KB / 2) * LDB_P];

    const int tid  = threadIdx.x;
    const int lane = tid & 31;
    const int wave = tid >> 5;
    const int blockM = blockIdx.x * BM;

    const int mr = lane & 15;           // M (A) / N (B) index within tile
    const int kh = (lane >> 4) << 1;    // K sub-offset within 4: 0 or 2

    v8f c[NT];
#pragma unroll
    for (int t = 0; t < NT; ++t) c[t] = (v8f){};

    for (int kb = 0; kb < K; kb += KB) {
        __syncthreads();
        // ---- stage A chunk: BM x KB, coalesced float4 along K ----
        {
            constexpr int F4 = BM * KB / 4;     // 1024 float4 elems
            for (int idx = tid; idx < F4; idx += 256) {
                const int row = idx >> 3;              // / (KB/4)
                const int kc  = (idx & 7) << 2;        // * 4
                float4 v;
                const int grow = blockM + row;
                if (grow < M) {
                    v = *(const float4*)(A + (size_t)grow * lda + kb + kc);
                } else {
                    v = make_float4(0.f, 0.f, 0.f, 0.f);
                }
                *(float4*)(&As[row * LDA_S + kc]) = v;
            }
        }
        // ---- stage B chunk: KB x BN, coalesced float4 load, paired-K store ----
        {
            constexpr int F4  = KB * BN / 4;
            constexpr int NB4 = BN / 4;
            for (int idx = tid; idx < F4; idx += 256) {
                const int row = idx / NB4;             // k within chunk
                const int col = (idx % NB4) << 2;      // n
                const float4 v = *(const float4*)(B + (size_t)(kb + row) * ldb + col);
                float* __restrict__ bp = &BsP[(row >> 1) * LDB_P + col * 2 + (row & 1)];
                bp[0] = v.x; bp[2] = v.y; bp[4] = v.z; bp[6] = v.w;
            }
        }
        __syncthreads();

        // ---- compute: 8 K-steps of 4, NT wmma per step ----
        const float* __restrict__ Aw = &As[(wave * 16 + mr) * LDA_S];
#pragma unroll 2
        for (int k0 = 0; k0 < KB; k0 += 4) {
            const v2f a = *(const v2f*)(&Aw[k0 + kh]);               // ds_load_b64
            const float* __restrict__ B0 = &BsP[((k0 + kh) >> 1) * LDB_P + mr * 2];
#pragma unroll
            for (int t = 0; t < NT; ++t) {
                const v2f b = *(const v2f*)(&B0[t * 32]);            // ds_load_b64
                c[t] = __builtin_amdgcn_wmma_f32_16x16x4_f32(
                        false, a, false, b, (short)0, c[t], false, false);
            }
        }
    }

    // ---- store: wave tile rows, guarded against M tail ----
    const int mbase = blockM + wave * 16 + ((lane >> 4) << 3);
#pragma unroll
    for (int t = 0; t < NT; ++t) {
        float* __restrict__ Cp = C + c_col_off + t * 16 + mr;
#pragma unroll
        for (int r = 0; r < 8; ++r) {
            const int row = mbase + r;
            if (row < M) Cp[(size_t)row * ldc] = c[t][r];
        }
    }
}

// ---------------------------------------------------------------------------
// Layer-1 edge aggregation: one wave per edge, 128 channels = 32 lanes x float4.
// agg[dst, :] += h[src, :] * dinv[src]*dinv[dst]
// ---------------------------------------------------------------------------
__global__ __launch_bounds__(256)
void edge_agg1_kernel(const float* __restrict__ h, const float* __restrict__ dinv,
                      const int* __restrict__ src, const int* __restrict__ dst,
                      float* __restrict__ agg, int nE) {
    const int e = blockIdx.x * 8 + (threadIdx.x >> 5);
    if (e >= nE) return;                       // uniform per wave, no barriers used
    const int lane = threadIdx.x & 31;
    const int s = __builtin_amdgcn_readfirstlane(src[e]);
    const int d = __builtin_amdgcn_readfirstlane(dst[e]);
    const float norm = dinv[s] * dinv[d];

    const int c4 = lane << 2;
    const float4 v = *(const float4*)(h + (size_t)s * HID_CH + c4);
    float* __restrict__ ap = agg + (size_t)d * HID_CH + c4;
    atomicAdd(ap + 0, v.x * norm);
    atomicAdd(ap + 1, v.y * norm);
    atomicAdd(ap + 2, v.z * norm);
    atomicAdd(ap + 3, v.w * norm);
}

// ---------------------------------------------------------------------------
// Layer-2 edge aggregation: one wave per edge, lin2 row = [mu(64) | ls(64)].
// Lanes 0-15 scatter to out_mu, lanes 16-31 to out_ls (no barrier, safe split).
// ---------------------------------------------------------------------------
__global__ __launch_bounds__(256)
void edge_agg2_kernel(const float* __restrict__ lin2, const float* __restrict__ dinv,
                      const int* __restrict__ src, const int* __restrict__ dst,
                      float* __restrict__ out, int nE) {
    const int e = blockIdx.x * 8 + (threadIdx.x >> 5);
    if (e >= nE) return;
    const int lane = threadIdx.x & 31;
    const int s = __builtin_amdgcn_readfirstlane(src[e]);
    const int d = __builtin_amdgcn_readfirstlane(dst[e]);
    const float norm = dinv[s] * dinv[d];

    const int c4 = lane << 2;                  // 0..124 across lin2's 128 cols
    const float4 v = *(const float4*)(lin2 + (size_t)s * 128 + c4);
    float* __restrict__ ap = (lane < 16)
        ? out + (size_t)d * OUT_CH + c4
        : out + (size_t)(N_NODES + d) * OUT_CH + (c4 - 64);
    atomicAdd(ap + 0, v.x * norm);
    atomicAdd(ap + 1, v.y * norm);
    atomicAdd(ap + 2, v.z * norm);
    atomicAdd(ap + 3, v.w * norm);
}

// ---------------------------------------------------------------------------
// Layer-1 combine: h = relu(agg + dinv^2 * hlin + b1), in place into hlin
// ---------------------------------------------------------------------------
__global__ __launch_bounds__(256)
void combine1_kernel(float* __restrict__ hlin, const float* __restrict__ agg,
                     const float* __restrict__ dinv, const float* __restrict__ b1) {
    const int idx = blockIdx.x * 256 + threadIdx.x;
    if (idx >= N_NODES * HID_CH) return;
    const int i  = idx >> 7;
    const int ch = idx & 127;
    float dii = dinv[i];
    dii *= dii;
    float v = agg[idx] + dii * hlin[idx] + b1[ch];
    hlin[idx] = v > 0.0f ? v : 0.0f;
}

// ---------------------------------------------------------------------------
// Layer-2 combine: out (holds edge agg) += dinv^2 * lin2 + bias
// ---------------------------------------------------------------------------
__global__ __launch_bounds__(256)
void combine2_kernel(const float* __restrict__ lin2, const float* __restrict__ dinv,
                     const float* __restrict__ b_mu, const float* __restrict__ b_ls,
                     float* __restrict__ out) {
    const int idx = blockIdx.x * 256 + threadIdx.x;
    if (idx >= N_NODES * 128) return;
    const int i  = idx >> 7;
    const int ch = idx & 127;
    float dii = dinv[i];
    dii *= dii;
    const float v = dii * lin2[idx];
    if (ch < 64) {
        size_t o = (size_t)i * OUT_CH + ch;
        out[o] = out[o] + v + b_mu[ch];
    } else {
        size_t o = (size_t)(N_NODES + i) * OUT_CH + (ch - 64);
        out[o] = out[o] + v + b_ls[ch - 64];
    }
}

// ---------------------------------------------------------------------------
extern "C" void kernel_launch(void* const* d_in, const int* in_sizes, int n_in,
                              void* d_out, int out_size, void* d_ws, size_t ws_size,
                              hipStream_t stream) {
    const float* x    = (const float*)d_in[0];
    const int*   edge = (const int*)d_in[1];
    const int*   src  = edge;
    const int*   dst  = edge + N_EDGES;
    const float* W1   = (const float*)d_in[2];
    const float* b1   = (const float*)d_in[3];
    const float* Wmu  = (const float*)d_in[4];
    const float* bmu  = (const float*)d_in[5];
    const float* Wls  = (const float*)d_in[6];
    const float* bls  = (const float*)d_in[7];
    float* out = (float*)d_out;

    // Workspace carve-up
    float* dinv = (float*)d_ws;                         // N_NODES
    float* hlin = dinv + N_NODES;                       // N_NODES*128
    float* agg1 = hlin + (size_t)N_NODES * HID_CH;      // N_NODES*128
    float* lin2 = agg1 + (size_t)N_NODES * HID_CH;      // N_NODES*128

    // Zero accumulators (graph-capturable async memsets)
    hipMemsetAsync(dinv, 0, (size_t)N_NODES * sizeof(float), stream);
    hipMemsetAsync(agg1, 0, (size_t)N_NODES * HID_CH * sizeof(float), stream);
    hipMemsetAsync(d_out, 0, (size_t)N_NODES * 2 * OUT_CH * sizeof(float), stream);

    // Degree + rsqrt
    deg_count_kernel<<<(N_EDGES + 255) / 256, 256, 0, stream>>>(dst, dinv, N_EDGES);
    rsqrt_kernel<<<(N_NODES + 255) / 256, 256, 0, stream>>>(dinv, N_NODES);

    const int mblocks = (N_NODES + 127) / 128;   // 391

    // GEMM1: hlin = x @ W1   (M=50000, N=128, K=256)
    gemm_wmma_f32_kernel<8><<<mblocks, 256, 0, stream>>>(
        x, W1, hlin, N_NODES, IN_CH, IN_CH, HID_CH, HID_CH, 0);

    // Layer-1 edge aggregation: one wave per edge
    edge_agg1_kernel<<<(N_EDGES + 7) / 8, 256, 0, stream>>>(
        hlin, dinv, src, dst, agg1, N_EDGES);

    combine1_kernel<<<(N_NODES * HID_CH + 255) / 256, 256, 0, stream>>>(
        hlin, agg1, dinv, b1);

    // GEMM2: lin2[:, :64] = h @ Wmu ; lin2[:, 64:] = h @ Wls  (N=64, K=128)
    gemm_wmma_f32_kernel<4><<<mblocks, 256, 0, stream>>>(
        hlin, Wmu, lin2, N_NODES, HID_CH, HID_CH, OUT_CH, HID_CH, 0);
    gemm_wmma_f32_kernel<4><<<mblocks, 256, 0, stream>>>(
        hlin, Wls, lin2, N_NODES, HID_CH, HID_CH, OUT_CH, HID_CH, 64);

    // Layer-2 edge aggregation into d_out: one wave per edge, both halves
    edge_agg2_kernel<<<(N_EDGES + 7) / 8, 256, 0, stream>>>(
        lin2, dinv, src, dst, out, N_EDGES);

    combine2_kernel<<<(N_NODES * 128 + 255) / 256, 256, 0, stream>>>(
        lin2, dinv, bmu, bls, out);
}